// Gate_38122129719677
// MI455X (gfx1250) — compile-verified
//
#include <hip/hip_runtime.h>
#include <hip/hip_bf16.h>
#include <math.h>

typedef __attribute__((ext_vector_type(2))) float v2f;
typedef __attribute__((ext_vector_type(8))) float v8f;

#define SE_B   64
#define SE_C   512
#define SE_CR  32
#define SE_HW  3136          // 56*56
#define SE_HW4 784           // 3136/4
#define SE_PLANES (SE_B * SE_C)   // 32768

// ---------------------------------------------------------------------------
// Kernel A: per-(b,c) plane mean.  One wave32 per plane, 8 planes per block.
// ---------------------------------------------------------------------------
__global__ void se_reduce_kernel(const float* __restrict__ x,
                                 float* __restrict__ s) {
    const int wave = threadIdx.x >> 5;
    const int lane = threadIdx.x & 31;
    const int plane = blockIdx.x * 8 + wave;           // 0..32767 exactly

    const float4* px = (const float4*)(x + (size_t)plane * SE_HW);
    float sum = 0.0f;
    for (int i = lane; i < SE_HW4; i += 32) {
        float4 v = px[i];
        sum += (v.x + v.y) + (v.z + v.w);
    }
    // wave32 tree reduction
    #pragma unroll
    for (int off = 16; off > 0; off >>= 1)
        sum += __shfl_down(sum, off, 32);
    if (lane == 0)
        s[plane] = sum * (1.0f / (float)SE_HW);
}

// ---------------------------------------------------------------------------
// Kernel B: gate computation with V_WMMA_F32_16X16X4_F32.
//   h[64,32]  = relu(s[64,512] @ w1[32,512]^T + b1)
//   g[64,512] = sigmoid(h[64,32] @ w2[512,32]^T + b2)
// Single 256-thread block = 8 waves; h staged in LDS between the two GEMMs.
//
// A 16x4 layout (32-bit): lane m = lane&15, kb = (lane>=16)?2:0,
//   vgpr0 = A[m][kb], vgpr1 = A[m][kb+1].  B 4x16 mirrors with n = lane&15.
// C/D 16x16: vgpr j -> row j + (lane>=16)*8, col = lane&15.
// ---------------------------------------------------------------------------
__global__ void se_gate_kernel(const float* __restrict__ s,
                               const float* __restrict__ w1,
                               const float* __restrict__ b1,
                               const float* __restrict__ w2,
                               const float* __restrict__ b2,
                               float* __restrict__ g) {
    __shared__ float h_lds[SE_B * SE_CR];              // 8 KB

    const int wave  = threadIdx.x >> 5;
    const int lane  = threadIdx.x & 31;
    const int mrow  = lane & 15;                       // M (and N) index within tile
    const int kb    = (lane >> 4) * 2;                 // K sub-offset for A/B frags
    const int rowhi = (lane >> 4) * 8;                 // C/D row offset for lanes 16-31

    // ---- GEMM1: 4 x 2 tiles of 16x16, one tile per wave -------------------
    {
        const int mt = wave >> 1;                      // 0..3  (B tiles)
        const int nt = wave & 1;                       // 0..1  (Cr tiles)
        const float* arow = s  + (size_t)(mt * 16 + mrow) * SE_C;
        const float* brow = w1 + (size_t)(nt * 16 + mrow) * SE_C;

        v8f acc = {0.f, 0.f, 0.f, 0.f, 0.f, 0.f, 0.f, 0.f};
        for (int k0 = 0; k0 < SE_C; k0 += 4) {
            v2f a = { arow[k0 + kb], arow[k0 + kb + 1] };
            v2f b = { brow[k0 + kb], brow[k0 + kb + 1] };
            acc = __builtin_amdgcn_wmma_f32_16x16x4_f32(
                false, a, false, b, (short)0, acc, false, false);
        }

        const int ncol = nt * 16 + (lane & 15);
        const float bias = b1[ncol];
        #pragma unroll
        for (int j = 0; j < 8; ++j) {
            const int mr = mt * 16 + j + rowhi;
            h_lds[mr * SE_CR + ncol] = fmaxf(acc[j] + bias, 0.0f);
        }
    }
    __syncthreads();

    // ---- GEMM2: 4 x 32 tiles of 16x16, 16 tiles per wave ------------------
    for (int t = wave; t < 128; t += 8) {              // wave-uniform loop
        const int mt = t >> 5;                         // 0..3  (B tiles)
        const int nt = t & 31;                         // 0..31 (C tiles)
        const float* arow = h_lds + (size_t)(mt * 16 + mrow) * SE_CR;
        const float* brow = w2    + (size_t)(nt * 16 + mrow) * SE_CR;

        v8f acc = {0.f, 0.f, 0.f, 0.f, 0.f, 0.f, 0.f, 0.f};
        #pragma unroll
        for (int k0 = 0; k0 < SE_CR; k0 += 4) {
            v2f a = { arow[k0 + kb], arow[k0 + kb + 1] };
            v2f b = { brow[k0 + kb], brow[k0 + kb + 1] };
            acc = __builtin_amdgcn_wmma_f32_16x16x4_f32(
                false, a, false, b, (short)0, acc, false, false);
        }

        const int ncol = nt * 16 + (lane & 15);
        const float bias = b2[ncol];
        #pragma unroll
        for (int j = 0; j < 8; ++j) {
            const int mr = mt * 16 + j + rowhi;
            const float v = acc[j] + bias;
            g[mr * SE_C + ncol] = 1.0f / (1.0f + __expf(-v));
        }
    }
}

// ---------------------------------------------------------------------------
// Kernel C: out = x * g[b,c].  One block per plane; gate is block-uniform
// (scalar load), float4 streaming.
// ---------------------------------------------------------------------------
__global__ void se_scale_kernel(const float* __restrict__ x,
                                const float* __restrict__ g,
                                float* __restrict__ out) {
    const int plane = blockIdx.x;
    const float gate = g[plane];                       // uniform -> s_load
    const float4* px = (const float4*)(x   + (size_t)plane * SE_HW);
    float4*       po = (float4*)      (out + (size_t)plane * SE_HW);
    for (int i = threadIdx.x; i < SE_HW4; i += 256) {
        float4 v = px[i];
        v.x *= gate; v.y *= gate; v.z *= gate; v.w *= gate;
        po[i] = v;
    }
}

extern "C" void kernel_launch(void* const* d_in, const int* in_sizes, int n_in,
                              void* d_out, int out_size, void* d_ws, size_t ws_size,
                              hipStream_t stream) {
    const float* x  = (const float*)d_in[0];
    const float* w1 = (const float*)d_in[1];
    const float* b1 = (const float*)d_in[2];
    const float* w2 = (const float*)d_in[3];
    const float* b2 = (const float*)d_in[4];
    float* out = (float*)d_out;

    float* s = (float*)d_ws;                 // [64*512] = 128 KB
    float* g = s + SE_PLANES;                // [64*512] = 128 KB

    se_reduce_kernel<<<SE_PLANES / 8, 256, 0, stream>>>(x, s);
    se_gate_kernel<<<1, 256, 0, stream>>>(s, w1, b1, w2, b2, g);
    se_scale_kernel<<<SE_PLANES, 256, 0, stream>>>(x, g, out);
}